// Tree_83915071030022
// MI455X (gfx1250) — compile-verified
//
#include <hip/hip_runtime.h>
#include <math.h>

typedef __attribute__((ext_vector_type(2))) float v2f;
typedef __attribute__((ext_vector_type(8))) float v8f;

#define WAVES 8   // 8 wave32 per block, 16 docs per wave -> 128 docs / block

__device__ __forceinline__ float fast_rcp(float x)  { return __builtin_amdgcn_rcpf(x); }
__device__ __forceinline__ float fast_sqrt(float x) { return __builtin_amdgcn_sqrtf(x); }

__global__ __launch_bounds__(256) void tree_dtd_kernel(
    const float* __restrict__ doc, const float* __restrict__ topic,
    float* __restrict__ out, int N)
{
  __shared__ float lds[WAVES][16][17];   // stride 17 avoids bank conflicts
  const int lane = threadIdx.x & 31;
  const int wave = threadIdx.x >> 5;
  const int lo   = lane & 15;
  const int hi   = lane >> 4;
  const long tile  = (long)blockIdx.x * WAVES + wave;
  const long mbase = tile * 16;

  // ---------- B tile: topic^T in WMMA-B layout (lane=col n, K = 4k + 2*hi + {0,1})
  v2f b[16];
  float y2p = 0.f;
#pragma unroll
  for (int k = 0; k < 16; ++k) {
    v2f t = {0.f, 0.f};
    if (lo < 13) t = *(const v2f*)(topic + lo * 64 + k * 4 + hi * 2);
    b[k] = t;
    y2p = fmaf(t.x, t.x, fmaf(t.y, t.y, y2p));
  }
  const float y2 = y2p + __shfl_xor(y2p, 16, 32);   // ||topic[lo]||^2 (0 for lo>=13)

  // ---------- A tile: issue ALL 16 loads first (16 outstanding b64 loads),
  // then run the serial WMMA accumulation chain.
  long row = mbase + lo;
  if (row >= N) row = N - 1;                        // safety clamp (N % 128 == 0 normally)
  const float* arow = doc + row * 64 + hi * 2;

  v2f a[16];
#pragma unroll
  for (int k = 0; k < 16; ++k) a[k] = *(const v2f*)(arow + k * 4);

  v8f acc = {0.f, 0.f, 0.f, 0.f, 0.f, 0.f, 0.f, 0.f};
  float x2p = 0.f;
#pragma unroll
  for (int k = 0; k < 16; ++k) {
    x2p = fmaf(a[k].x, a[k].x, fmaf(a[k].y, a[k].y, x2p));
    // full-precision fp32 matrix op: v_wmma_f32_16x16x4_f32
    acc = __builtin_amdgcn_wmma_f32_16x16x4_f32(false, a[k], false, b[k],
                                                (short)0, acc, false, false);
  }
  const float x2l = x2p + __shfl_xor(x2p, 16, 32);  // ||doc[mbase+lo]||^2

  // ---------- Poincare sqdist -> eta, transposed into LDS
#pragma unroll
  for (int j = 0; j < 8; ++j) {
    const int m = hi * 8 + j;                       // C layout: VGPR j -> row 8*hi+j
    const float xy = acc[j];
    const float x2 = __shfl(x2l, m, 32);            // fetch x2 of row m
    const float A = 1.f - 2.f * xy + y2;
    const float B = 1.f - x2;
    const float num2  = A * A * x2 - 2.f * A * B * xy + B * B * y2;
    const float denom = fmaxf(fmaf(x2, y2, 1.f - 2.f * xy), 1e-15f);
    const float mn = fast_sqrt(fmaxf(num2, 0.f) + 1e-15f) * fast_rcp(denom);
    const float t  = fminf(mn, 1.f - 1e-7f);
    // dist = 2*artanh(t) = ln((1+t)/(1-t)); t in ~[0.1,0.6] here -> no cancellation
    const float dist = 0.69314718056f *
                       __builtin_amdgcn_logf((1.f + t) * fast_rcp(1.f - t));
    const float dd = dist * dist;
    lds[wave][m][lo] = fast_rcp(1.f + dd);          // eta
  }
  __syncthreads();

  // ---------- tree epilogue: lanes 0-15 each own one doc row
  if (hi == 0) {
    float e[13];
#pragma unroll
    for (int t = 0; t < 13; ++t) e[t] = lds[wave][lo][t];

    // level distribution (min-dist topic == max-eta topic)
    float m1 = fmaxf(fmaxf(e[1], e[2]), e[3]);
    float m2 = e[4];
#pragma unroll
    for (int t = 5; t < 13; ++t) m2 = fmaxf(m2, e[t]);
    const float rs = fast_rcp(e[0] + m1 + m2 + 1e-20f);
    const float L0 = e[0] * rs, L1 = m1 * rs, L2 = m2 * rs;

    // gamma within sibling groups; collapse path-scatter analytically
    const float rS1 = fast_rcp(e[1] + e[2] + e[3] + 1e-20f);
    float o[13];
    float total = 0.f;
#pragma unroll
    for (int p = 0; p < 3; ++p) {
      const float gp = e[1 + p] * rS1;
      const float c0 = e[4 + 3 * p], c1 = e[5 + 3 * p], c2 = e[6 + 3 * p];
      const float rSp = fast_rcp(c0 + c1 + c2 + 1e-20f);
      const float h0 = c0 * rSp, h1 = c1 * rSp, h2 = c2 * rSp;
      const float branch = gp * (h0 + h1 + h2);
      total += branch;
      o[1 + p]     = L1 * branch;
      o[4 + 3 * p] = L2 * gp * h0;
      o[5 + 3 * p] = L2 * gp * h1;
      o[6 + 3 * p] = L2 * gp * h2;
    }
    o[0] = L0 * total;
#pragma unroll
    for (int t = 0; t < 13; ++t) lds[wave][lo][t] = o[t];
  }
  __syncthreads();

  // ---------- coalesced store: 16 docs * 13 = 208 contiguous floats per wave
  float* obase = out + mbase * 13;
  const long lim = (long)N * 13 - mbase * 13;
  for (int idx = lane; idx < 208; idx += 32) {
    if (idx < lim) obase[idx] = lds[wave][idx / 13][idx % 13];
  }
}

extern "C" void kernel_launch(void* const* d_in, const int* in_sizes, int n_in,
                              void* d_out, int out_size, void* d_ws, size_t ws_size,
                              hipStream_t stream) {
  const float* doc   = (const float*)d_in[0];   // [N,64] f32
  const float* topic = (const float*)d_in[1];   // [13,64] f32
  float* out = (float*)d_out;                   // [N,13] f32
  const int N = in_sizes[0] / 64;
  const int docs_per_block = 16 * WAVES;
  const int grid = (N + docs_per_block - 1) / docs_per_block;
  tree_dtd_kernel<<<grid, 256, 0, stream>>>(doc, topic, out, N);
}